// LevelMapper_76776835383906
// MI455X (gfx1250) — compile-verified
//
#include <hip/hip_runtime.h>
#include <hip/hip_bf16.h>
#include <math.h>

typedef float v2f __attribute__((ext_vector_type(2)));
typedef float v4f __attribute__((ext_vector_type(4)));
typedef float v8f __attribute__((ext_vector_type(8)));

#define N_ROIS 64
#define CCH    64
#define OUTW   128

// ---------------------------------------------------------------------------
// Kernel 1: build composed bilinear (align_corners) upsample matrices.
// W0 = I(16x16); W_l (S x 16), S = 16*2^l, l=1..3. Row-major f32 in ws.
// Offsets (floats): level offset = 256*((1<<level)-1); total 3840 floats.
// ---------------------------------------------------------------------------
__global__ __launch_bounds__(256) void build_weights_kernel(float* __restrict__ wbuf) {
    __shared__ float prev[2048];
    __shared__ float cur[2048];
    const int tid = threadIdx.x;

    // W0 = identity 16x16
    {
        float v = ((tid >> 4) == (tid & 15)) ? 1.0f : 0.0f;
        prev[tid] = v;
        wbuf[tid] = v;
    }
    __syncthreads();

    int off = 256;
    for (int l = 1; l <= 3; ++l) {
        const int n     = 16 << (l - 1);   // input rows of this upsample step
        const int rows  = 2 * n;           // output rows
        const int total = rows * 16;
        const float step = (float)(n - 1) / (float)(2 * n - 1);
        for (int i = tid; i < total; i += 256) {
            const int j = i >> 4;
            const int c = i & 15;
            const float src = (float)j * step;
            int i0 = (int)floorf(src);
            int i1 = min(i0 + 1, n - 1);
            const float t = src - (float)i0;
            cur[i] = (1.0f - t) * prev[i0 * 16 + c] + t * prev[i1 * 16 + c];
        }
        __syncthreads();
        for (int i = tid; i < total; i += 256) {
            wbuf[off + i] = cur[i];
            prev[i] = cur[i];
        }
        off += total;
        __syncthreads();
    }
}

// ---------------------------------------------------------------------------
// Kernel 2: one block per (roi, channel). out = M_l * crop(16x16) * M_l^T,
// placed top-left of 128x128, zeros elsewhere. Uses V_WMMA_F32_16X16X4_F32.
// ---------------------------------------------------------------------------
__global__ __launch_bounds__(256) void roi_level_wmma_kernel(
    const float* __restrict__ x0, const float* __restrict__ x1,
    const float* __restrict__ x2, const float* __restrict__ x3,
    const float* __restrict__ rois, const float* __restrict__ wbuf,
    float* __restrict__ out)
{
    __shared__ float crop_s[16 * 16];
    __shared__ float T_s[128 * 16];

    const int blk  = blockIdx.x;
    const int roi  = blk >> 6;      // / CCH
    const int ch   = blk & 63;      // % CCH
    const int tid  = threadIdx.x;
    const int lane = tid & 31;
    const int wv   = tid >> 5;      // wave id 0..7 (wave32)
    const int half = lane >> 4;     // 0: lanes 0-15, 1: lanes 16-31
    const int ln   = lane & 15;

    // --- ROI -> level / crop origin (reference semantics) ---
    const float px = rois[roi * 5 + 1];
    const float py = rois[roi * 5 + 2];
    const float wR = rois[roi * 5 + 3];
    const float hR = rois[roi * 5 + 4];
    const int   b  = (int)rois[roi * 5 + 0];

    const float s   = sqrtf(wR * hR);
    float tgt = floorf(4.0f + log2f(s / 224.0f + 1e-6f));
    tgt = fminf(fmaxf(tgt, 2.0f), 5.0f);
    const int level = (int)tgt - 2;               // 0..3

    const float scale = 0.25f / (float)(1 << level);
    const int xl = ((int)(px * scale)) / 2 * 2;
    const int yl = ((int)(py * scale)) / 2 * 2;
    const int Hl = 256 >> level;                  // feature H == W
    const float* feat = (level == 0) ? x0 : (level == 1) ? x1 : (level == 2) ? x2 : x3;
    const float* src  = feat + (((size_t)(b * CCH + ch)) * Hl + yl) * (size_t)Hl + xl;

    const int R = 1 << level;        // 16x16 tiles per side
    const int S = 16 << level;       // upsampled patch size
    const float* Wl = wbuf + 256 * ((1 << level) - 1);   // M_l, (R*16) x 16 row-major
    float* img = out + (size_t)blk * (OUTW * OUTW);

    // --- load 16x16 crop into LDS (one element per thread) ---
    crop_s[tid] = src[(size_t)(tid >> 4) * Hl + (tid & 15)];
    __syncthreads();

    // --- stage 1: T = M_l * crop  (each wave: one 16-row tile) ---
    if (wv < R) {
        const int r = wv;
        v8f acc = {};
        #pragma unroll
        for (int kk = 0; kk < 4; ++kk) {
            v2f a, bb;
            // A = M_l rows r*16.., cols kk*4..kk*4+3 (16x4, ISA lane layout)
            a.x = Wl[(r * 16 + ln) * 16 + kk * 4 + half * 2 + 0];
            a.y = Wl[(r * 16 + ln) * 16 + kk * 4 + half * 2 + 1];
            // B = crop rows kk*4..kk*4+3 (4x16)
            bb.x = crop_s[(kk * 4 + half * 2 + 0) * 16 + ln];
            bb.y = crop_s[(kk * 4 + half * 2 + 1) * 16 + ln];
            acc = __builtin_amdgcn_wmma_f32_16x16x4_f32(
                false, a, false, bb, (short)0, acc, false, false);
        }
        // D layout: VGPR v, lane-half h -> row v + 8h, col ln
        #pragma unroll
        for (int v = 0; v < 8; ++v)
            T_s[(r * 16 + v + half * 8) * 16 + ln] = acc[v];
    }
    __syncthreads();

    // --- stage 2: O_{r,cb} = T_r * (M_l^T)_cb ; store 16x16 tiles ---
    for (int t = wv; t < R * R; t += 8) {
        const int r  = t / R;
        const int cb = t - r * R;
        v8f acc = {};
        #pragma unroll
        for (int kk = 0; kk < 4; ++kk) {
            v2f a, bb;
            a.x = T_s[(r * 16 + ln) * 16 + kk * 4 + half * 2 + 0];
            a.y = T_s[(r * 16 + ln) * 16 + kk * 4 + half * 2 + 1];
            // B[k][n] = (M^T)[k][cb*16+n] = M[cb*16+n][k]
            bb.x = Wl[(cb * 16 + ln) * 16 + kk * 4 + half * 2 + 0];
            bb.y = Wl[(cb * 16 + ln) * 16 + kk * 4 + half * 2 + 1];
            acc = __builtin_amdgcn_wmma_f32_16x16x4_f32(
                false, a, false, bb, (short)0, acc, false, false);
        }
        #pragma unroll
        for (int v = 0; v < 8; ++v) {
            const int row = r * 16 + v + half * 8;
            const int col = cb * 16 + ln;
            __builtin_nontemporal_store(acc[v], img + row * OUTW + col);
        }
    }

    // --- zero-fill outside S x S (S multiple of 16 -> float4 chunks uniform) ---
    const v4f z = {0.f, 0.f, 0.f, 0.f};
    for (int i = tid; i < (OUTW * OUTW) / 4; i += 256) {
        const int y  = i >> 5;            // (i*4)/128
        const int x4 = (i & 31) << 2;     // (i*4)%128
        if (y >= S || x4 >= S)
            __builtin_nontemporal_store(z, reinterpret_cast<v4f*>(img) + i);
    }
}

// ---------------------------------------------------------------------------
extern "C" void kernel_launch(void* const* d_in, const int* in_sizes, int n_in,
                              void* d_out, int out_size, void* d_ws, size_t ws_size,
                              hipStream_t stream) {
    const float* x0   = (const float*)d_in[0];
    const float* x1   = (const float*)d_in[1];
    const float* x2   = (const float*)d_in[2];
    const float* x3   = (const float*)d_in[3];
    const float* rois = (const float*)d_in[4];
    float* out = (float*)d_out;
    float* wbuf = (float*)d_ws;   // 3840 floats used

    build_weights_kernel<<<1, 256, 0, stream>>>(wbuf);
    roi_level_wmma_kernel<<<N_ROIS * CCH, 256, 0, stream>>>(
        x0, x1, x2, x3, rois, wbuf, out);
}